// GCNConvolution_88055419503314
// MI455X (gfx1250) — compile-verified
//
#include <hip/hip_runtime.h>

#define N_NODES 50000
#define F_IN 256
#define F_OUT 10
#define F_OUT_PAD 16
#define MTILES (N_NODES / 16)   // 3125 exactly

typedef float v2f __attribute__((ext_vector_type(2)));
typedef float v8f __attribute__((ext_vector_type(8)));

// ---------------- degree / normalization ----------------

__global__ void deg_kernel(const long long* __restrict__ dst, float* __restrict__ deg, int E) {
    int e = blockIdx.x * blockDim.x + threadIdx.x;
    if (e < E) atomicAdd(&deg[dst[e]], 1.0f);
}

__global__ void dinv_kernel(const float* __restrict__ deg, float* __restrict__ dinv, int n) {
    int i = blockIdx.x * blockDim.x + threadIdx.x;
    if (i < n) dinv[i] = rsqrtf(deg[i] + 1.0f);
}

// ---------------- GEMM1: h1 = x @ W1  (fp32 WMMA 16x16x4) ----------------

__global__ void __launch_bounds__(128)
gemm1_wmma(const float* __restrict__ X, const float* __restrict__ W, float* __restrict__ H) {
    const int wave  = threadIdx.x >> 5;
    const int lane  = threadIdx.x & 31;
    const int mtile = blockIdx.x * 4 + wave;
    if (mtile >= MTILES) return;                 // wave-uniform exit
    const int ntile = blockIdx.y;
    const int row0  = mtile * 16;
    const int col0  = ntile * 16;
    const int lmod  = lane & 15;
    const int khalf = (lane >> 4) << 1;          // 0 for lanes 0-15, 2 for lanes 16-31

    const float* __restrict__ arow = X + (size_t)(row0 + lmod) * F_IN + khalf;
    const float* __restrict__ bcol = W + (size_t)khalf * F_IN + col0 + lmod;

    v8f acc = {};
    for (int k = 0; k < F_IN; k += 4) {
        v2f a; a.x = arow[k];     a.y = arow[k + 1];            // A: rows of x
        v2f b; b.x = bcol[(size_t)k * F_IN];                    // B: rows of W1
               b.y = bcol[(size_t)(k + 1) * F_IN];
        acc = __builtin_amdgcn_wmma_f32_16x16x4_f32(
            false, a, false, b, (short)0, acc, false, false);
    }
    const int crow = row0 + (lane >> 4) * 8;
    float* __restrict__ out = H + (size_t)crow * F_IN + col0 + lmod;
#pragma unroll
    for (int v = 0; v < 8; ++v) out[(size_t)v * F_IN] = acc[v];
}

// ---------------- self-loop init: agg1 = h1 * dinv^2 ----------------

__global__ void self_init1(const float* __restrict__ H, const float* __restrict__ dinv,
                           float* __restrict__ agg) {
    size_t i = (size_t)blockIdx.x * blockDim.x + threadIdx.x;   // over N*F_IN/4
    if (i >= (size_t)N_NODES * (F_IN / 4)) return;
    size_t node = i / (F_IN / 4);
    float di = dinv[node];
    float s = di * di;
    float4 v = ((const float4*)H)[i];
    float4 o; o.x = v.x * s; o.y = v.y * s; o.z = v.z * s; o.w = v.w * s;
    ((float4*)agg)[i] = o;
}

// ---------------- edge scatter (layer 1): agg1[dst] += h1[src]*norm ----------------

__global__ void scatter1(const long long* __restrict__ src, const long long* __restrict__ dst,
                         const float* __restrict__ dinv, const float* __restrict__ H,
                         float* __restrict__ agg, int E) {
    size_t tid = (size_t)blockIdx.x * blockDim.x + threadIdx.x;
    if (tid >= (size_t)E * 64) return;
    size_t e = tid >> 6;
    int c = (int)(tid & 63) << 2;
    long long s = src[e], d = dst[e];
    float norm = dinv[s] * dinv[d];
    float4 v = *(const float4*)(H + (size_t)s * F_IN + c);      // gather (L2-resident)
    float* o = agg + (size_t)d * F_IN + c;
    atomicAdd(o + 0, v.x * norm);
    atomicAdd(o + 1, v.y * norm);
    atomicAdd(o + 2, v.z * norm);
    atomicAdd(o + 3, v.w * norm);
}

// ---------------- GEMM2: h2 = relu(agg1 + b1) @ W2  (WMMA, N padded to 16) ----------------

__global__ void __launch_bounds__(128)
gemm2_wmma(const float* __restrict__ A, const float* __restrict__ b1,
           const float* __restrict__ W2, float* __restrict__ H2) {
    const int wave  = threadIdx.x >> 5;
    const int lane  = threadIdx.x & 31;
    const int mtile = blockIdx.x * 4 + wave;
    if (mtile >= MTILES) return;
    const int row0  = mtile * 16;
    const int lmod  = lane & 15;
    const int khalf = (lane >> 4) << 1;
    // column mask instead of divergent branch: EXEC must stay all-1s for WMMA
    const int   col   = (lmod < F_OUT) ? lmod : (F_OUT - 1);
    const float bmask = (lmod < F_OUT) ? 1.0f : 0.0f;

    const float* __restrict__ arow = A + (size_t)(row0 + lmod) * F_IN + khalf;

    v8f acc = {};
    for (int k = 0; k < F_IN; k += 4) {
        v2f a;                                                   // fused bias + relu
        a.x = fmaxf(arow[k]     + b1[k + khalf],     0.0f);
        a.y = fmaxf(arow[k + 1] + b1[k + khalf + 1], 0.0f);
        v2f b;
        b.x = W2[(size_t)(k + khalf)     * F_OUT + col] * bmask;
        b.y = W2[(size_t)(k + khalf + 1) * F_OUT + col] * bmask;
        acc = __builtin_amdgcn_wmma_f32_16x16x4_f32(
            false, a, false, b, (short)0, acc, false, false);
    }
    const int crow = row0 + (lane >> 4) * 8;
    float* __restrict__ out = H2 + (size_t)crow * F_OUT_PAD + lmod;
#pragma unroll
    for (int v = 0; v < 8; ++v) out[(size_t)v * F_OUT_PAD] = acc[v];
}

// ---------------- output init: out = h2*dinv^2 + b2 ----------------

__global__ void out_init(const float* __restrict__ H2, const float* __restrict__ dinv,
                         const float* __restrict__ b2, float* __restrict__ out) {
    int i = blockIdx.x * blockDim.x + threadIdx.x;
    if (i >= N_NODES * F_OUT) return;
    int n = i / F_OUT, c = i - n * F_OUT;
    float di = dinv[n];
    out[i] = H2[(size_t)n * F_OUT_PAD + c] * di * di + b2[c];
}

// ---------------- edge scatter (layer 2): out[dst] += h2[src]*norm ----------------

__global__ void scatter2(const long long* __restrict__ src, const long long* __restrict__ dst,
                         const float* __restrict__ dinv, const float* __restrict__ H2,
                         float* __restrict__ out, int E) {
    size_t tid = (size_t)blockIdx.x * blockDim.x + threadIdx.x;
    if (tid >= (size_t)E * 16) return;
    size_t e = tid >> 4;
    int c = (int)(tid & 15);
    if (c >= F_OUT) return;
    long long s = src[e], d = dst[e];
    float norm = dinv[s] * dinv[d];
    atomicAdd(out + (size_t)d * F_OUT + c, H2[(size_t)s * F_OUT_PAD + c] * norm);
}

// ---------------- launch ----------------

extern "C" void kernel_launch(void* const* d_in, const int* in_sizes, int n_in,
                              void* d_out, int out_size, void* d_ws, size_t ws_size,
                              hipStream_t stream) {
    const float*     x    = (const float*)d_in[0];
    const long long* ei   = (const long long*)d_in[1];   // [2, E] int64
    const float*     W1   = (const float*)d_in[2];
    const float*     b1   = (const float*)d_in[3];
    const float*     W2   = (const float*)d_in[4];
    const float*     b2   = (const float*)d_in[5];
    float*           out  = (float*)d_out;

    const int E = in_sizes[1] / 2;
    const long long* src  = ei;
    const long long* dstp = ei + E;

    // workspace layout (floats): deg[50048] | dinv[50048] | h1[12.8M] | agg1[12.8M]
    float* ws   = (float*)d_ws;
    float* deg  = ws;
    float* dinv = ws + 50048;
    float* h1   = ws + 100096;
    float* agg1 = h1 + (size_t)N_NODES * F_IN;
    float* h2   = h1;                       // reuse h1 buffer after scatter1 is done

    // 1. degree + dinv (deg must be re-zeroed every call: deterministic launch)
    hipMemsetAsync(deg, 0, N_NODES * sizeof(float), stream);
    deg_kernel<<<(E + 255) / 256, 256, 0, stream>>>(dstp, deg, E);
    dinv_kernel<<<(N_NODES + 255) / 256, 256, 0, stream>>>(deg, dinv, N_NODES);

    // 2. h1 = x @ W1 (WMMA fp32)
    dim3 g1((MTILES + 3) / 4, F_IN / 16);
    gemm1_wmma<<<g1, 128, 0, stream>>>(x, W1, h1);

    // 3. agg1 = h1 * dinv^2  (also fully initializes agg1 buffer)
    size_t n4 = (size_t)N_NODES * (F_IN / 4);
    self_init1<<<(unsigned)((n4 + 255) / 256), 256, 0, stream>>>(h1, dinv, agg1);

    // 4. scatter over edges (layer 1)
    size_t t1 = (size_t)E * 64;
    scatter1<<<(unsigned)((t1 + 255) / 256), 256, 0, stream>>>(src, dstp, dinv, h1, agg1, E);

    // 5. h2 = relu(agg1 + b1) @ W2 (WMMA, padded N=16)
    gemm2_wmma<<<(MTILES + 3) / 4, 128, 0, stream>>>(agg1, b1, W2, h2);

    // 6. out = h2*dinv^2 + b2, then scatter (layer 2)
    out_init<<<(N_NODES * F_OUT + 255) / 256, 256, 0, stream>>>(h2, dinv, b2, out);
    size_t t2 = (size_t)E * 16;
    scatter2<<<(unsigned)((t2 + 255) / 256), 256, 0, stream>>>(src, dstp, dinv, h2, out, E);

    // 7. tuple return: append edge_index (raw int64 bytes) after the [50000,10] floats
    hipMemcpyAsync((char*)d_out + (size_t)N_NODES * F_OUT * sizeof(float),
                   (const void*)ei, (size_t)in_sizes[1] * sizeof(long long),
                   hipMemcpyDeviceToDevice, stream);
}